// Embedder_71408126264020
// MI455X (gfx1250) — compile-verified
//
#include <hip/hip_runtime.h>
#include <stdint.h>

// Problem constants (from reference): G=16, B=8192, K=1024, D=64
#define G_ 16
#define B_ 8192
#define K_ 1024
#define D_ 64

#define BM 256      // x rows per workgroup tile (32 per wave -> 2 M-tiles/wave)
#define KC 256      // centroids resident in LDS per chunk (double-buffered)
#define SP 68       // padded LDS row stride in floats (bank-conflict-free: n*68 % 64 = 4n)
#define NCHUNK (K_ / KC)

typedef __attribute__((ext_vector_type(2))) float v2f;
typedef __attribute__((ext_vector_type(8))) float v8f;

// LDS aperture lives entirely in addr[63:32]; low 32 bits = wave-relative LDS offset.
__device__ __forceinline__ uint32_t lds_off(const void* p) {
    return (uint32_t)(uintptr_t)p;
}

// Async copy of one KC x D f32 centroid chunk into padded LDS rows.
// 256 threads x 16 iters x b128 = 64 KB payload, tracked on ASYNCcnt.
__device__ __forceinline__ void async_copy_chunk(float* dst, const float* src_base, int tid) {
    #pragma unroll
    for (int r = 0; r < (KC * D_) / (256 * 4); ++r) {   // 16 iterations
        int i   = tid + r * 256;
        int row = i >> 4;                               // 16 float4 per row
        int c4  = i & 15;
        uint32_t lds  = lds_off(&dst[row * SP + c4 * 4]);
        uint32_t voff = (uint32_t)(row * (D_ * 4) + c4 * 16);  // byte offset from base
        asm volatile("global_load_async_to_lds_b128 %0, %1, %2"
                     :: "v"(lds), "v"(voff), "s"(src_base)
                     : "memory");
    }
}

__device__ __forceinline__ void wait_async_all() {
    asm volatile("s_wait_asynccnt 0x0" ::: "memory");
}

__global__ __launch_bounds__(256)
void dist_wmma_f32_kernel(const float* __restrict__ x,
                          const float* __restrict__ cen,
                          float* __restrict__ out) {
    extern __shared__ float smem[];
    float* xs  = smem;                       // BM * SP
    float* cs0 = xs + BM * SP;               // KC * SP  (buffer 0)
    float* cs1 = cs0 + KC * SP;              // KC * SP  (buffer 1)
    float* x2  = cs1 + KC * SP;              // BM
    float* c2  = x2 + BM;                    // KC

    const int tid  = threadIdx.x;
    const int lane = tid & 31;
    const int wave = tid >> 5;               // 0..7, each wave owns 32 rows
    const int g    = blockIdx.y;
    const int b0   = blockIdx.x * BM;

    const float* cen_g = cen + (size_t)g * K_ * D_;

    // ---- kick off async load of centroid chunk 0 while we stage the x tile ----
    async_copy_chunk(cs0, cen_g, tid);

    // ---- load x tile [BM x D] into LDS, coalesced float4 ----
    {
        const float4* src = (const float4*)(x + (size_t)(g * B_ + b0) * D_);
        #pragma unroll
        for (int r = 0; r < (BM * D_ / 4) / 256; ++r) {
            int i   = tid + r * 256;
            int row = i >> 4;
            int c4  = i & 15;
            float4 v = src[row * (D_ / 4) + c4];
            *(float4*)&xs[row * SP + c4 * 4] = v;
        }
    }
    __syncthreads();                         // xs visible to all waves

    // ---- row norms ||x||^2 (one row per thread; BM == blockDim) ----
    {
        float s = 0.f;
        #pragma unroll
        for (int d = 0; d < D_; ++d) { float t = xs[tid * SP + d]; s += t * t; }
        x2[tid] = s;
    }

    // ---- A fragments for BOTH M-tiles, held in registers for the whole kernel ----
    // V_WMMA_F32_16X16X4_F32 A layout (16x4 MxK): lanes 0-15: K={0,1}; lanes 16-31: K={2,3}
    const int mrow0 = (wave << 5) + (lane & 15);   // tile 0 row
    const int mrow1 = mrow0 + 16;                  // tile 1 row
    const int koff  = (lane >> 4) << 1;            // 0 or 2
    v2f afrag0[D_ / 4], afrag1[D_ / 4];
    #pragma unroll
    for (int kk = 0; kk < D_ / 4; ++kk) {
        afrag0[kk] = *(const v2f*)&xs[mrow0 * SP + kk * 4 + koff];
        afrag1[kk] = *(const v2f*)&xs[mrow1 * SP + kk * 4 + koff];
    }
    __syncthreads();                         // x2 visible

    // x2 per-lane registers matching C/D layout: M = j + 8*(lane>=16)
    float x2r0[8], x2r1[8];
    #pragma unroll
    for (int j = 0; j < 8; ++j) {
        int ml = j + ((lane >> 4) << 3);
        x2r0[j] = x2[(wave << 5) + ml];
        x2r1[j] = x2[(wave << 5) + 16 + ml];
    }

    const int ncol = lane & 15;

    // ---- double-buffered centroid chunks: async prefetch overlaps WMMA ----
    int cur = 0;
    for (int cc = 0; cc < NCHUNK; ++cc) {
        wait_async_all();                    // this wave's async loads landed
        __syncthreads();                     // all waves' loads landed; prev compute done

        float* cs = cur ? cs1 : cs0;
        if (cc + 1 < NCHUNK) {               // prefetch next chunk into other buffer
            float* csn = cur ? cs0 : cs1;
            async_copy_chunk(csn, cen_g + (size_t)(cc + 1) * KC * D_, tid);
        }

        // centroid norms for this chunk (one row per thread; KC == blockDim)
        {
            float s = 0.f;
            #pragma unroll
            for (int d = 0; d < D_; ++d) { float t = cs[tid * SP + d]; s += t * t; }
            c2[tid] = s;
        }
        __syncthreads();

        const int k0 = cc * KC;

        // ---- WMMA: 32 rows x KC cols per wave; B fragment shared by 2 accumulators ----
        #pragma unroll 2
        for (int nt = 0; nt < KC / 16; ++nt) {
            const int n     = nt * 16 + ncol;
            const float c2v = c2[n];
            v8f acc0 = {}, acc1 = {};
            #pragma unroll
            for (int kk = 0; kk < D_ / 4; ++kk) {
                // B layout (4x16 KxN), symmetric K split with A: lanes 0-15: K={0,1}, 16-31: K={2,3}
                v2f bfrag = *(const v2f*)&cs[n * SP + kk * 4 + koff];
                acc0 = __builtin_amdgcn_wmma_f32_16x16x4_f32(
                    false, afrag0[kk], false, bfrag, (short)0, acc0, false, false);
                acc1 = __builtin_amdgcn_wmma_f32_16x16x4_f32(
                    false, afrag1[kk], false, bfrag, (short)0, acc1, false, false);
            }
            #pragma unroll
            for (int j = 0; j < 8; ++j) {
                int   ml = j + ((lane >> 4) << 3);
                size_t base = (size_t)(g * B_ + b0 + (wave << 5) + ml) * K_ + (k0 + n);
                float d2a = fmaxf(x2r0[j] + c2v - 2.0f * acc0[j], 0.0f);
                float d2b = fmaxf(x2r1[j] + c2v - 2.0f * acc1[j], 0.0f);
                // streaming output: non-temporal stores keep centroids L2-resident
                __builtin_nontemporal_store(sqrtf(d2a), &out[base]);
                __builtin_nontemporal_store(sqrtf(d2b), &out[base + (size_t)16 * K_]);
            }
        }
        cur ^= 1;
    }
}

extern "C" void kernel_launch(void* const* d_in, const int* in_sizes, int n_in,
                              void* d_out, int out_size, void* d_ws, size_t ws_size,
                              hipStream_t stream) {
    (void)in_sizes; (void)n_in; (void)out_size; (void)d_ws; (void)ws_size;
    const float* x   = (const float*)d_in[0];
    const float* cen = (const float*)d_in[1];
    float* out = (float*)d_out;

    dim3 grid(B_ / BM, G_);
    dim3 block(256);
    size_t shmem = (size_t)(BM * SP + 2 * KC * SP + BM + KC) * sizeof(float); // ~206 KB
    dist_wmma_f32_kernel<<<grid, block, shmem, stream>>>(x, cen, out);
}